// GMMflow_64450279244471
// MI455X (gfx1250) — compile-verified
//
#include <hip/hip_runtime.h>
#include <math.h>

// ---------------------------------------------------------------------------
// GMM flow refactored into two fp32 WMMA GEMMs (V_WMMA_F32_16X16X4_F32).
//   logw[b,p] = [X^2 | X | 1](b,:) . G(:,p)        (GEMM1, K=144)
//   W = exp(clip(logw))*Lam                         (fused epilogue)
//   R[b,:]   = W(b,:) . [K | C | 1]                 (GEMM2, K=1024)
//   out[b,n] = (X[b,n]*R[b,n] + R[b,64+n]) / R[b,128]
// ---------------------------------------------------------------------------

typedef float v2f __attribute__((ext_vector_type(2)));
typedef float v8f __attribute__((ext_vector_type(8)));

constexpr int Bsz = 1024;              // batch
constexpr int Dn  = 64;                // feature dim
constexpr int Np  = 1024;              // 32*32 pairs
constexpr int K1  = 144;               // padded inner dim of GEMM1 (2*64+1 -> 144)
constexpr int Qn  = 144;               // padded output cols of GEMM2 (2*64+1 -> 144)

#define EPS2f 0.25f
#define EPS4f 0.0625f

// ---- fp32 WMMA fragment loaders (layouts per CDNA5 ISA 7.12.2) -------------
// A 16x4: lane L(<16) holds row M=L, K=k0+0/1 in v.x/v.y; lane L+16 holds K=k0+2/3.
__device__ inline v2f load_a_frag(const float* __restrict__ A, int lda,
                                  int m0, int k0, int lane) {
    int row = m0 + (lane & 15);
    int kk  = k0 + ((lane >> 4) << 1);
    const float* p = A + row * lda + kk;
    v2f a; a.x = p[0]; a.y = p[1];
    return a;
}
// B 4x16: lane L(<16) holds col N=n0+L, rows K=k0+0/1 in v.x/v.y; lane L+16 rows k0+2/3.
__device__ inline v2f load_b_frag(const float* __restrict__ Bm, int ldb,
                                  int k0, int n0, int lane) {
    int col = n0 + (lane & 15);
    int kr  = k0 + ((lane >> 4) << 1);
    v2f b; b.x = Bm[kr * ldb + col]; b.y = Bm[(kr + 1) * ldb + col];
    return b;
}

__device__ inline v8f wmma_f32(v2f a, v2f b, v8f c) {
    return __builtin_amdgcn_wmma_f32_16x16x4_f32(
        /*neg_a=*/false, a, /*neg_b=*/false, b,
        /*c_mod=*/(short)0, c, /*reuse_a=*/false, /*reuse_b=*/false);
}

// ---- Kernel 1: per-pair parameters ----------------------------------------
// block = one pair p (64 threads = feature dims). Fills G (144 x 1024) and
// H (1024 x 144) plus the per-pair constant (LDS reduction).
__global__ void GMMflow_pair_setup(const float* __restrict__ Mu0,
                                   const float* __restrict__ Mu1,
                                   const float* __restrict__ S0,
                                   const float* __restrict__ S1,
                                   const float* __restrict__ tptr,
                                   float* __restrict__ G,
                                   float* __restrict__ H) {
    const int p = blockIdx.x;          // pair index = i*32 + j
    const int i = p >> 5, j = p & 31;
    const int d = threadIdx.x;         // 0..63
    const float t  = *tptr;
    const float mt = 1.0f - t;

    const float s0  = S0[i * Dn + d];
    const float s1  = S1[j * Dn + d];
    const float mu0 = Mu0[i * Dn + d];
    const float mu1 = Mu1[j * Dn + d];

    const float Ds    = sqrtf(4.0f * s0 * s1 + EPS4f);
    const float Cs    = 0.5f * (Ds - EPS2f);
    const float mu    = mt * mu0 + t * mu1;
    const float Sigma = mt * mt * s0 + t * t * s1 + 2.0f * t * mt * (Cs + 0.5f * EPS2f);
    const float St    = t * s1 + mt * Cs - (mt * s0 + t * Cs) - EPS2f * t;
    const float Kc    = St / Sigma;
    const float v     = mu1 - mu0;
    const float Cc    = v - Kc * mu;
    const float invS  = 1.0f / Sigma;

    // GEMM1 B-matrix rows: X^2 coeff, X coeff; pad rows zeroed.
    G[d * Np + p]        = -0.5f * invS;
    G[(Dn + d) * Np + p] = mu * invS;
    if (d < 15) G[(129 + d) * Np + p] = 0.0f;

    // GEMM2 B-matrix: [K | C | 1 | 0-pad] per pair row.
    H[p * Qn + d]      = Kc;
    H[p * Qn + Dn + d] = Cc;
    if (d == 0) H[p * Qn + 128] = 1.0f;
    if (d < 15) H[p * Qn + 129 + d] = 0.0f;

    // per-pair constant: -0.5 * sum_d (mu^2*invS + log Sigma)
    __shared__ float sh[Dn];
    sh[d] = mu * mu * invS + logf(Sigma);
    __syncthreads();
    if (d == 0) {
        float acc = 0.0f;
        #pragma unroll
        for (int k = 0; k < Dn; ++k) acc += sh[k];
        G[128 * Np + p] = -0.5f * acc;
    }
}

// ---- Kernel 2: build Xt = [X^2 | X | 1 | 0-pad]  (1024 x 144) --------------
__global__ void GMMflow_build_xt(const float* __restrict__ X,
                                 float* __restrict__ Xt) {
    const int b = blockIdx.x;
    const int k = threadIdx.x;         // 0..143
    float v;
    if (k < Dn)            { float x = X[b * Dn + k];      v = x * x; }
    else if (k < 2 * Dn)   { v = X[b * Dn + (k - Dn)]; }
    else if (k == 2 * Dn)  { v = 1.0f; }
    else                   { v = 0.0f; }
    Xt[b * K1 + k] = v;
}

// ---- Kernel 3: GEMM1 (1024x1024, K=144) + fused exp epilogue ---------------
// block = 256 threads = 8 waves as 4(M) x 2(N); each wave does 32x64 output.
__global__ void GMMflow_gemm_logw(const float* __restrict__ Xt,
                                  const float* __restrict__ G,
                                  const float* __restrict__ Lam,
                                  float* __restrict__ Wbuf) {
    const int lane = threadIdx.x & 31;
    const int wid  = threadIdx.x >> 5;
    const int wm   = wid & 3;
    const int wn   = wid >> 2;
    const int m_base = blockIdx.y * 128 + wm * 32;
    const int n_base = blockIdx.x * 128 + wn * 64;

    v8f acc[2][4];
    const v8f zero = {0.f, 0.f, 0.f, 0.f, 0.f, 0.f, 0.f, 0.f};
    #pragma unroll
    for (int a = 0; a < 2; ++a)
        #pragma unroll
        for (int b = 0; b < 4; ++b) acc[a][b] = zero;

    for (int k0 = 0; k0 < K1; k0 += 4) {
        v2f af[2], bf[4];
        #pragma unroll
        for (int a = 0; a < 2; ++a)
            af[a] = load_a_frag(Xt, K1, m_base + 16 * a, k0, lane);
        #pragma unroll
        for (int b = 0; b < 4; ++b)
            bf[b] = load_b_frag(G, Np, k0, n_base + 16 * b, lane);
        #pragma unroll
        for (int a = 0; a < 2; ++a)
            #pragma unroll
            for (int b = 0; b < 4; ++b)
                acc[a][b] = wmma_f32(af[a], bf[b], acc[a][b]);
    }

    // Epilogue: W = exp(clip(logw,-50,50)) * Lam[col]
    const int cn = lane & 15;
    const int rh = (lane >> 4) << 3;   // 0 or 8
    #pragma unroll
    for (int b = 0; b < 4; ++b) {
        const int col = n_base + 16 * b + cn;
        const float lam = Lam[col];
        #pragma unroll
        for (int a = 0; a < 2; ++a) {
            const int row0 = m_base + 16 * a + rh;
            v8f c = acc[a][b];
            #pragma unroll
            for (int r = 0; r < 8; ++r) {
                float lw = fminf(fmaxf(c[r], -50.0f), 50.0f);
                Wbuf[(row0 + r) * Np + col] = __expf(lw) * lam;
            }
        }
    }
}

// ---- Kernel 4: GEMM2 (1024x144, K=1024): R = W . [K|C|1] -------------------
// block = 256 threads = 8 waves stacked in M; each wave does 32x48 output.
__global__ void GMMflow_gemm_num(const float* __restrict__ Wbuf,
                                 const float* __restrict__ H,
                                 float* __restrict__ R) {
    const int lane = threadIdx.x & 31;
    const int wid  = threadIdx.x >> 5;
    const int m_base = blockIdx.y * 256 + wid * 32;
    const int n_base = blockIdx.x * 48;

    v8f acc[2][3];
    const v8f zero = {0.f, 0.f, 0.f, 0.f, 0.f, 0.f, 0.f, 0.f};
    #pragma unroll
    for (int a = 0; a < 2; ++a)
        #pragma unroll
        for (int b = 0; b < 3; ++b) acc[a][b] = zero;

    for (int k0 = 0; k0 < Np; k0 += 4) {
        v2f af[2], bf[3];
        #pragma unroll
        for (int a = 0; a < 2; ++a)
            af[a] = load_a_frag(Wbuf, Np, m_base + 16 * a, k0, lane);
        #pragma unroll
        for (int b = 0; b < 3; ++b)
            bf[b] = load_b_frag(H, Qn, k0, n_base + 16 * b, lane);
        #pragma unroll
        for (int a = 0; a < 2; ++a)
            #pragma unroll
            for (int b = 0; b < 3; ++b)
                acc[a][b] = wmma_f32(af[a], bf[b], acc[a][b]);
    }

    const int cn = lane & 15;
    const int rh = (lane >> 4) << 3;
    #pragma unroll
    for (int b = 0; b < 3; ++b) {
        const int col = n_base + 16 * b + cn;
        #pragma unroll
        for (int a = 0; a < 2; ++a) {
            const int row0 = m_base + 16 * a + rh;
            v8f c = acc[a][b];
            #pragma unroll
            for (int r = 0; r < 8; ++r)
                R[(row0 + r) * Qn + col] = c[r];
        }
    }
}

// ---- Kernel 5: finalize out = (X*R1 + R2) / den ----------------------------
__global__ void GMMflow_finalize(const float* __restrict__ X,
                                 const float* __restrict__ R,
                                 float* __restrict__ out) {
    const int idx = blockIdx.x * blockDim.x + threadIdx.x;   // B*Dn
    const int b = idx >> 6, nn = idx & 63;
    const float num = X[idx] * R[b * Qn + nn] + R[b * Qn + Dn + nn];
    const float den = R[b * Qn + 128];
    out[idx] = num / den;
}

// ---------------------------------------------------------------------------
extern "C" void kernel_launch(void* const* d_in, const int* in_sizes, int n_in,
                              void* d_out, int out_size, void* d_ws, size_t ws_size,
                              hipStream_t stream) {
    const float* X   = (const float*)d_in[0];
    const float* Mu0 = (const float*)d_in[1];
    const float* Mu1 = (const float*)d_in[2];
    const float* S0  = (const float*)d_in[3];
    const float* S1  = (const float*)d_in[4];
    const float* Lam = (const float*)d_in[5];
    const float* t   = (const float*)d_in[6];
    float* out = (float*)d_out;

    // workspace carve-up (floats): total 1,638,400 floats = 6.25 MB
    float* ws   = (float*)d_ws;
    float* G    = ws;                       // 144 * 1024
    float* Xt   = G  + K1 * Np;             // 1024 * 144
    float* H    = Xt + Bsz * K1;            // 1024 * 144
    float* R    = H  + Np * Qn;             // 1024 * 144
    float* Wbuf = R  + Bsz * Qn;            // 1024 * 1024

    GMMflow_pair_setup<<<Np, Dn, 0, stream>>>(Mu0, Mu1, S0, S1, t, G, H);
    GMMflow_build_xt<<<Bsz, K1, 0, stream>>>(X, Xt);
    GMMflow_gemm_logw<<<dim3(Np / 128, Bsz / 128), 256, 0, stream>>>(Xt, G, Lam, Wbuf);
    GMMflow_gemm_num<<<dim3(Qn / 48, Bsz / 256), 256, 0, stream>>>(Wbuf, H, R);
    GMMflow_finalize<<<(Bsz * Dn) / 256, 256, 0, stream>>>(X, R, out);
}